// ChebGNN_26242250179009
// MI455X (gfx1250) — compile-verified
//
#include <hip/hip_runtime.h>

#define N_NODES 50000
#define N_EDGES 800000
#define CH      128      // IN_C == HID
#define K3      384      // 3 * CH (fused Chebyshev K dimension)
#define OUTC    2

typedef __attribute__((ext_vector_type(2))) float v2f;
typedef __attribute__((ext_vector_type(8))) float v8f;

// ---------------- utility kernels ----------------
__global__ void fill_zero_kernel(float* __restrict__ p, int n) {
  int i = blockIdx.x * blockDim.x + threadIdx.x;
  if (i < n) p[i] = 0.0f;
}

__global__ void deg_kernel(const int* __restrict__ row, float* __restrict__ deg, int nE) {
  int e = blockIdx.x * blockDim.x + threadIdx.x;
  if (e < nE) atomicAdd(&deg[row[e]], 1.0f);
}

__global__ void dinv_kernel(float* __restrict__ d, int n) {
  int i = blockIdx.x * blockDim.x + threadIdx.x;
  if (i < n) {
    float v = d[i];
    d[i] = (v > 0.0f) ? rsqrtf(v) : 0.0f;
  }
}

__global__ void wgt_kernel(const int* __restrict__ row, const int* __restrict__ col,
                           const float* __restrict__ dinv, float* __restrict__ w, int nE) {
  int e = blockIdx.x * blockDim.x + threadIdx.x;
  if (e < nE) w[e] = -dinv[row[e]] * dinv[col[e]];
}

__global__ void neg_copy_kernel(float* __restrict__ dst, const float* __restrict__ src, int n) {
  int i = blockIdx.x * blockDim.x + threadIdx.x;
  if (i < n) dst[i] = -src[i];
}

// P[col[e], :] += scale * w[e] * T[row[e], :]   (32 lanes/edge, 4 channels/lane)
__global__ void scatter_kernel(const float* __restrict__ T, const int* __restrict__ row,
                               const int* __restrict__ col, const float* __restrict__ w,
                               float* __restrict__ P, float scale, int nE) {
  int idx  = blockIdx.x * blockDim.x + threadIdx.x;
  int e    = idx >> 5;
  int lane = idx & 31;
  if (e >= nE) return;
  float wv = w[e] * scale;
  int r = row[e];
  int c = col[e];
  float4 v = *reinterpret_cast<const float4*>(T + (size_t)r * CH + lane * 4);
  float* dst = P + (size_t)c * CH + lane * 4;
  atomicAdd(dst + 0, wv * v.x);
  atomicAdd(dst + 1, wv * v.y);
  atomicAdd(dst + 2, wv * v.z);
  atomicAdd(dst + 3, wv * v.w);
}

// ---------------- fused Chebyshev GEMM: Out = relu([A0|A1|A2] @ W(384x128) + bias) ----------------
#define ASTRIDE 33    // 128x32 A tile, pad to 33 floats -> conflict-free strided reads
#define BSTRIDE 132   // 32x128 B tile, pad to 132 floats (16B-aligned rows for float4 stores)
#define TK      32

__global__ __launch_bounds__(256)
void cheb_gemm_relu_kernel(const float* __restrict__ A0, const float* __restrict__ A1,
                           const float* __restrict__ A2, const float* __restrict__ W,
                           const float* __restrict__ bias, float* __restrict__ Out,
                           int nRows) {
  __shared__ float As[128 * ASTRIDE];
  __shared__ float Bs[TK * BSTRIDE];

  const int tid  = threadIdx.x;
  const int wave = tid >> 5;   // 8 waves (wave32)
  const int lane = tid & 31;
  const int m0   = blockIdx.x * 128;

  v8f acc[8];
#pragma unroll
  for (int i = 0; i < 8; ++i) acc[i] = 0.0f;

  const float* Abuf[3] = {A0, A1, A2};

  for (int kc = 0; kc < K3; kc += TK) {
    const float* Ab = Abuf[kc >> 7];       // each 32-chunk stays inside one Tx buffer
    const int kl = kc & (CH - 1);

    // cooperative load: A tile 128x32 (row-clamped so EXEC stays uniform)
#pragma unroll
    for (int i = 0; i < 4; ++i) {
      int idx  = tid + i * 256;            // 0..1023 float4 slots
      int rowi = idx >> 3;
      int c4   = (idx & 7) * 4;
      int gm   = m0 + rowi;
      if (gm > nRows - 1) gm = nRows - 1;
      float4 v = *reinterpret_cast<const float4*>(Ab + (size_t)gm * CH + kl + c4);
      float* d = &As[rowi * ASTRIDE + c4];
      d[0] = v.x; d[1] = v.y; d[2] = v.z; d[3] = v.w;
    }
    // cooperative load: B tile 32x128
#pragma unroll
    for (int i = 0; i < 4; ++i) {
      int idx  = tid + i * 256;
      int rowi = idx >> 5;
      int c4   = (idx & 31) * 4;
      float4 v = *reinterpret_cast<const float4*>(W + (size_t)(kc + rowi) * CH + c4);
      *reinterpret_cast<float4*>(&Bs[rowi * BSTRIDE + c4]) = v;
    }
    __syncthreads();

    const int am    = lane & 15;            // M (A) / N (B) index within tile
    const int khalf = (lane >> 4) << 1;     // lanes 16-31 hold K+2,K+3 halves
#pragma unroll
    for (int kk = 0; kk < TK; kk += 4) {
      v2f a;
      const float* ap = &As[(wave * 16 + am) * ASTRIDE + kk + khalf];
      a.x = ap[0];
      a.y = ap[1];
#pragma unroll
      for (int nt = 0; nt < 8; ++nt) {
        v2f b;
        const float* bp = &Bs[(kk + khalf) * BSTRIDE + nt * 16 + am];
        b.x = bp[0];
        b.y = bp[BSTRIDE];
        acc[nt] = __builtin_amdgcn_wmma_f32_16x16x4_f32(
            false, a, false, b, (short)0, acc[nt], false, false);
      }
    }
    __syncthreads();
  }

  // epilogue: C layout = VGPR v holds rows v (lanes 0-15) and v+8 (lanes 16-31)
  const int rb = (lane >> 4) * 8;
  const int cn = lane & 15;
  for (int nt = 0; nt < 8; ++nt) {
    int cidx = nt * 16 + cn;
    float bv = bias[cidx];
#pragma unroll
    for (int v = 0; v < 8; ++v) {
      int r = m0 + wave * 16 + rb + v;
      if (r < nRows) {
        float val = acc[nt][v] + bv;
        Out[(size_t)r * CH + cidx] = fmaxf(val, 0.0f);
      }
    }
  }
}

// ---------------- mean pool + linear head ----------------
__global__ void colsum_kernel(const float* __restrict__ h, float* __restrict__ colsum, int n) {
  int c = threadIdx.x;  // 128 threads
  float s = 0.0f;
  for (int r = blockIdx.x; r < n; r += gridDim.x) s += h[(size_t)r * CH + c];
  atomicAdd(&colsum[c], s);
}

__global__ void head_kernel(const float* __restrict__ colsum, const float* __restrict__ Wlin,
                            const float* __restrict__ blin, float* __restrict__ out) {
  int o = threadIdx.x;
  if (o < OUTC) {
    float s = blin[o];
    const float inv_n = 1.0f / (float)N_NODES;
    for (int c = 0; c < CH; ++c) s += (colsum[c] * inv_n) * Wlin[c * OUTC + o];
    out[o] = s;
  }
}

// ---------------- launch ----------------
extern "C" void kernel_launch(void* const* d_in, const int* in_sizes, int n_in,
                              void* d_out, int out_size, void* d_ws, size_t ws_size,
                              hipStream_t stream) {
  (void)in_sizes; (void)n_in; (void)out_size; (void)ws_size;

  const float* x    = (const float*)d_in[0];
  const int*   ei   = (const int*)d_in[1];
  const float* W1   = (const float*)d_in[2];   // (3,128,128) == (384,128) row-major
  const float* b1   = (const float*)d_in[3];
  const float* W2   = (const float*)d_in[4];
  const float* b2   = (const float*)d_in[5];
  const float* Wlin = (const float*)d_in[6];
  const float* blin = (const float*)d_in[7];
  float* out = (float*)d_out;

  const int* row = ei;              // edge_index[0]
  const int* col = ei + N_EDGES;    // edge_index[1]

  float* ws   = (float*)d_ws;
  float* dinv = ws;                                  // 50000 (deg, then dinv in place)
  float* wgt  = ws + N_NODES;                        // 800000
  float* Tx1  = wgt + N_EDGES;                       // 6.4M
  float* Tx2  = Tx1 + (size_t)N_NODES * CH;          // 6.4M
  float* h1   = Tx2 + (size_t)N_NODES * CH;          // 6.4M
  float* csum = h1  + (size_t)N_NODES * CH;          // 128
  float* h2   = out + OUTC;                          // layer-2 output h goes to d_out

  const int T = 256;
  const int NC = N_NODES * CH;
  const int scatT = N_EDGES * 32;
  auto cdiv = [](int a, int b) { return (a + b - 1) / b; };

  // edge normalization: deg -> dinv -> w
  fill_zero_kernel<<<cdiv(N_NODES, T), T, 0, stream>>>(dinv, N_NODES);
  deg_kernel<<<cdiv(N_EDGES, T), T, 0, stream>>>(row, dinv, N_EDGES);
  dinv_kernel<<<cdiv(N_NODES, T), T, 0, stream>>>(dinv, N_NODES);
  wgt_kernel<<<cdiv(N_EDGES, T), T, 0, stream>>>(row, col, dinv, wgt, N_EDGES);

  // ---- layer 1: h1 = relu([x | prop(x) | 2*prop(prop(x)) - x] @ W1 + b1)
  fill_zero_kernel<<<cdiv(NC, T), T, 0, stream>>>(Tx1, NC);
  scatter_kernel<<<cdiv(scatT, T), T, 0, stream>>>(x, row, col, wgt, Tx1, 1.0f, N_EDGES);
  neg_copy_kernel<<<cdiv(NC, T), T, 0, stream>>>(Tx2, x, NC);
  scatter_kernel<<<cdiv(scatT, T), T, 0, stream>>>(Tx1, row, col, wgt, Tx2, 2.0f, N_EDGES);
  cheb_gemm_relu_kernel<<<cdiv(N_NODES, 128), 256, 0, stream>>>(x, Tx1, Tx2, W1, b1, h1, N_NODES);

  // ---- layer 2: h2 = relu([h1 | prop(h1) | 2*prop(prop(h1)) - h1] @ W2 + b2), written into d_out
  fill_zero_kernel<<<cdiv(NC, T), T, 0, stream>>>(Tx1, NC);
  scatter_kernel<<<cdiv(scatT, T), T, 0, stream>>>(h1, row, col, wgt, Tx1, 1.0f, N_EDGES);
  neg_copy_kernel<<<cdiv(NC, T), T, 0, stream>>>(Tx2, h1, NC);
  scatter_kernel<<<cdiv(scatT, T), T, 0, stream>>>(Tx1, row, col, wgt, Tx2, 2.0f, N_EDGES);
  cheb_gemm_relu_kernel<<<cdiv(N_NODES, 128), 256, 0, stream>>>(h1, Tx1, Tx2, W2, b2, h2, N_NODES);

  // ---- mean pool + linear head
  fill_zero_kernel<<<1, 128, 0, stream>>>(csum, CH);
  colsum_kernel<<<512, 128, 0, stream>>>(h2, csum, N_NODES);
  head_kernel<<<1, 32, 0, stream>>>(csum, Wlin, blin, out);
}